// LSTMLayer_64424509440733
// MI455X (gfx1250) — compile-verified
//
#include <hip/hip_runtime.h>
#include <hip/hip_bf16.h>
#include <stdint.h>

typedef __attribute__((ext_vector_type(2))) float v2f;
typedef __attribute__((ext_vector_type(8))) float v8f;

// ---------------------------------------------------------------------------
// Fragment loaders for V_WMMA_F32_16X16X4_F32 (wave32).
// A is 16(M) x 4(K):  lanes 0-15 -> M=lane, VGPR0/1 = K0/K1; lanes 16-31 -> K2/K3.
// B is 4(K) x 16(N):  VGPR v, lanes 0-15 -> K=v, lanes 16-31 -> K=v+2; lane&15 = N.
// C is 16x16:         VGPR r, lanes 0-15 -> M=r, lanes 16-31 -> M=r+8; lane&15 = N.
// ---------------------------------------------------------------------------
__device__ __forceinline__ v2f load_a_frag(const float* __restrict__ A, int lda,
                                           int row0, int k0, int lane) {
  const int r  = lane & 15;
  const int kp = (lane >> 4) << 1;                 // 0 or 2
  const float* p = A + (size_t)(row0 + r) * lda + k0 + kp;
  return *(const v2f*)p;                            // K, K+1 contiguous
}

__device__ __forceinline__ v2f load_b_frag(const float* __restrict__ Bm, int ldb,
                                           int k0, int col0, int lane) {
  const int c  = lane & 15;
  const int kb = (lane >> 4) << 1;                 // 0 or 2
  const float* p = Bm + (size_t)(k0 + kb) * ldb + col0 + c;
  v2f b;
  b.x = p[0];                                       // row K
  b.y = p[ldb];                                     // row K+1
  return b;
}

__device__ __forceinline__ v8f wmma_f32(v2f a, v2f b, v8f c) {
  // (neg_a, A, neg_b, B, c_mod, C, reuse_a, reuse_b)
  return __builtin_amdgcn_wmma_f32_16x16x4_f32(false, a, false, b, (short)0, c,
                                               false, false);
}

__device__ __forceinline__ float sigmoid_f(float v) {
  return 1.0f / (1.0f + __expf(-v));
}

// CDNA5 async global->LDS copy (per-lane 16B), tracked by ASYNCcnt.
__device__ __forceinline__ void async_load_b128(uint32_t lds_addr,
                                                const float* gptr) {
  asm volatile("global_load_async_to_lds_b128 %0, %1, off"
               :
               : "v"(lds_addr), "v"((uint64_t)(uintptr_t)gptr)
               : "memory");
}
__device__ __forceinline__ void wait_asynccnt0() {
  asm volatile("s_wait_asynccnt 0x0" ::: "memory");
}

// ---------------------------------------------------------------------------
// Kernel A: xp[M,4096] = x[M,1024] @ W_in[1024,4096] + (b_in + b_h)
// Block = 256 threads (8 waves). Block tile = 64 rows x 128 cols.
// Wave w: rows (w&3)*16..+16, cols (w>>2)*64..+64 (4 accumulators).
// ---------------------------------------------------------------------------
__global__ __launch_bounds__(256) void lstm_xproj_kernel(
    const float* __restrict__ x, const float* __restrict__ W_in,
    const float* __restrict__ b_in, const float* __restrict__ b_h,
    float* __restrict__ xp, int M) {
  const int lane = threadIdx.x & 31;
  const int wave = threadIdx.x >> 5;
  const int row0 = blockIdx.x * 64 + (wave & 3) * 16;
  const int col0 = blockIdx.y * 128 + (wave >> 2) * 64;
  (void)M;

  v8f acc[4] = {};
  for (int k0 = 0; k0 < 1024; k0 += 4) {
    v2f a = load_a_frag(x, 1024, row0, k0, lane);
#pragma unroll
    for (int j = 0; j < 4; ++j) {
      v2f b = load_b_frag(W_in, 4096, k0, col0 + 16 * j, lane);
      acc[j] = wmma_f32(a, b, acc[j]);
    }
  }

  const int cc    = lane & 15;
  const int rbase = (lane >> 4) << 3;
#pragma unroll
  for (int j = 0; j < 4; ++j) {
    const int col    = col0 + 16 * j + cc;
    const float bias = b_in[col] + b_h[col];
#pragma unroll
    for (int r = 0; r < 8; ++r) {
      xp[(size_t)(row0 + rbase + r) * 4096 + col] = acc[j][r] + bias;
    }
  }
}

// ---------------------------------------------------------------------------
// Kernel B: one LSTM timestep.
//   gates[64,4096] = xp_t + h_prev @ W_h        (biases already in xp_t)
//   c = c*sig(f) + sig(i)*tanh(g);  h = tanh(c)*sig(o)
// Grid = (64 column-groups, 4 row-tiles), 128 threads (4 waves).
// Wave w computes gate chunk q=w's 16x16 tile with two interleaved WMMA
// accumulator chains over K. The xp tile is DMA'd to LDS with
// global_load_async_to_lds_b128 concurrently with the GEMM, drained by
// s_wait_asynccnt, and the gate nonlinearity fuses gates+xp from LDS.
// ---------------------------------------------------------------------------
#define LP 68  // LDS row pitch (floats): 272B rows, 16B aligned, 32-bank skew

__global__ __launch_bounds__(128) void lstm_step_kernel(
    const float* __restrict__ xp_t,    // [64, 4096]
    const float* __restrict__ h_prev,  // [64, 1024]
    const float* __restrict__ W_h,     // [1024, 4096]
    const float* __restrict__ c_in,    // [64, 1024]
    float* __restrict__ c_out,         // [64, 1024]
    float* __restrict__ h_out) {       // [64, 1024] == outputs[t]
  __shared__ float xpl[16 * LP];       // xp tile   [16 rows][64 gate-cols]
  __shared__ float gates[16 * LP];     // h@W_h tile[16 rows][64 gate-cols]

  const int lane = threadIdx.x & 31;
  const int q    = threadIdx.x >> 5;   // wave id == gate chunk (f,i,g,o)
  const int g16  = blockIdx.x * 16;    // H-column group
  const int m0   = blockIdx.y * 16;    // batch-row tile

  // Kick off async DMA of this wave's xp sub-tile (16 rows x 16 cols = 1KB)
  // while the GEMM runs. Per-lane B128 segments; 2 wave-instructions.
  {
    const uint32_t xpl_base = (uint32_t)(uintptr_t)&xpl[0];
    const float* gbase = xp_t + (size_t)m0 * 4096 + q * 1024 + g16;
#pragma unroll
    for (int i = 0; i < 2; ++i) {
      const int slot = i * 32 + lane;       // 0..63
      const int r = slot >> 2;              // row 0..15
      const int s = slot & 3;               // 16B quarter of the 64B row
      async_load_b128(xpl_base + (uint32_t)((r * LP + q * 16 + s * 4) * 4),
                      gbase + (size_t)r * 4096 + s * 4);
    }
  }

  // GEMM: two interleaved accumulator chains to hide WMMA latency.
  const int colB = q * 1024 + g16;
  v8f acc0 = {}, acc1 = {};
  for (int k0 = 0; k0 < 1024; k0 += 8) {
    v2f a0 = load_a_frag(h_prev, 1024, m0, k0, lane);
    v2f b0 = load_b_frag(W_h, 4096, k0, colB, lane);
    v2f a1 = load_a_frag(h_prev, 1024, m0, k0 + 4, lane);
    v2f b1 = load_b_frag(W_h, 4096, k0 + 4, colB, lane);
    acc0 = wmma_f32(a0, b0, acc0);
    acc1 = wmma_f32(a1, b1, acc1);
  }
  const v8f acc = acc0 + acc1;

  // Stash raw GEMM tile in LDS.
  const int cc    = lane & 15;
  const int rbase = (lane >> 4) << 3;
#pragma unroll
  for (int r = 0; r < 8; ++r) {
    gates[(rbase + r) * LP + q * 16 + cc] = acc[r];
  }

  wait_asynccnt0();   // xp DMA done (this wave's segments)
  __syncthreads();    // gates + all waves' xp visible block-wide

  // Fused gate nonlinearity: 16 rows x 16 cols, 2 elements per thread.
  const int row   = threadIdx.x & 15;
  const int jbase = (threadIdx.x >> 4) * 2;
  const int brow  = m0 + row;
#pragma unroll
  for (int jj = 0; jj < 2; ++jj) {
    const int jl = jbase + jj;
    const float f  = gates[row * LP + jl]      + xpl[row * LP + jl];
    const float ii = gates[row * LP + 16 + jl] + xpl[row * LP + 16 + jl];
    const float gg = gates[row * LP + 32 + jl] + xpl[row * LP + 32 + jl];
    const float oo = gates[row * LP + 48 + jl] + xpl[row * LP + 48 + jl];
    const int colH = g16 + jl;
    const float cprev = c_in[brow * 1024 + colH];
    const float cn = cprev * sigmoid_f(f) + sigmoid_f(ii) * tanhf(gg);
    const float hn = tanhf(cn) * sigmoid_f(oo);
    c_out[brow * 1024 + colH] = cn;
    h_out[brow * 1024 + colH] = hn;
  }
}

// ---------------------------------------------------------------------------
extern "C" void kernel_launch(void* const* d_in, const int* in_sizes, int n_in,
                              void* d_out, int out_size, void* d_ws,
                              size_t ws_size, hipStream_t stream) {
  (void)in_sizes; (void)n_in; (void)out_size;
  constexpr int T = 512, B = 64, D = 1024, H = 1024;

  const float* x    = (const float*)d_in[0];
  const float* h0   = (const float*)d_in[1];
  const float* c0   = (const float*)d_in[2];
  const float* W_in = (const float*)d_in[3];
  const float* b_in = (const float*)d_in[4];
  const float* W_h  = (const float*)d_in[5];
  const float* b_h  = (const float*)d_in[6];
  float* out = (float*)d_out;  // [T,B,H] ++ h_T[B,H] ++ c_T[B,H]

  const size_t c_bytes = (size_t)B * H * sizeof(float);
  float* cbuf  = (float*)d_ws;
  float* xpbuf = (float*)((char*)d_ws + c_bytes);

  // Adaptive time-chunking of the x-projection scratch (1 MB per timestep).
  const size_t per_t = (size_t)B * 4 * H * sizeof(float);
  const size_t avail = (ws_size > c_bytes) ? (ws_size - c_bytes) : 0;
  int Tc = (int)(avail / per_t);
  if (Tc > T) Tc = T;
  if (Tc < 1) Tc = 1;

  for (int t0 = 0; t0 < T; t0 += Tc) {
    const int tc = (T - t0 < Tc) ? (T - t0) : Tc;
    const int M  = tc * B;
    dim3 gA(M / 64, (4 * H) / 128);
    lstm_xproj_kernel<<<gA, 256, 0, stream>>>(x + (size_t)t0 * B * D, W_in,
                                              b_in, b_h, xpbuf, M);
    for (int t = t0; t < t0 + tc; ++t) {
      const float* hp = (t == 0) ? h0 : out + (size_t)(t - 1) * B * H;
      const float* ci = (t == 0) ? c0 : cbuf;
      lstm_step_kernel<<<dim3(64, 4), 128, 0, stream>>>(
          xpbuf + (size_t)(t - t0) * B * 4 * H, hp, W_h, ci, cbuf,
          out + (size_t)t * B * H);
    }
  }

  // Append h_T (= outputs[T-1]) and c_T.
  float* hT_dst = out + (size_t)T * B * H;
  float* cT_dst = hT_dst + (size_t)B * H;
  hipMemcpyAsync(hT_dst, out + (size_t)(T - 1) * B * H, c_bytes,
                 hipMemcpyDeviceToDevice, stream);
  hipMemcpyAsync(cT_dst, cbuf, c_bytes, hipMemcpyDeviceToDevice, stream);
}